// MemN2N_52390011077103
// MI455X (gfx1250) — compile-verified
//
#include <hip/hip_runtime.h>
#include <stdint.h>

// MemN2N attention, single-pass online-softmax streaming kernel for gfx1250.
// B=256 batches -> 256 workgroups; 512 threads (16 wave32) per workgroup.
// story[b] (2 MB) streamed ONCE through LDS with async global->LDS b128 copies
// (double buffered, ASYNCcnt). acc updated via V_WMMA_F32_16X16X4_F32.

#define D_       512
#define S_       1024
#define TS       16            // s-rows per tile
#define NT       (S_ / TS)     // 64 tiles
#define LSTRIDE  520           // padded LDS row stride (floats): kills bank conflicts
#define NTHREADS 512

typedef __attribute__((ext_vector_type(2))) float v2f;
typedef __attribute__((ext_vector_type(8))) float v8f;

#if defined(__HIP_DEVICE_COMPILE__) && __has_builtin(__builtin_amdgcn_wmma_f32_16x16x4_f32)
#define USE_WMMA 1
#else
#define USE_WMMA 0
#endif

// CDNA5 async copy: per-lane 16B global -> LDS, tracked by ASYNCcnt.
__device__ __forceinline__ void async_copy_b128(uint32_t lds_byte_off, const float* gsrc) {
  asm volatile("global_load_async_to_lds_b128 %0, %1, off"
               :: "v"(lds_byte_off), "v"(gsrc)
               : "memory");
}

template <int N>
__device__ __forceinline__ void wait_asynccnt() {
  asm volatile("s_wait_asynccnt %0" :: "i"(N) : "memory");
}

__global__ __launch_bounds__(NTHREADS) void memn2n_attn_kernel(
    const float* __restrict__ story,   // [B,S,D]
    const float* __restrict__ query,   // [B,D]
    const float* __restrict__ Wa_w,    // [D]
    const float* __restrict__ Wa_b,    // [1]
    float* __restrict__ out)           // [B,D]
{
  __shared__ __align__(16) float buf[2][TS * LSTRIDE];  // double-buffered story tile
  __shared__ float qs[D_];
  __shared__ float ws[D_];
  __shared__ float sc[TS];
  __shared__ __align__(16) float pb[TS];  // softmax weights for current tile
  __shared__ float scl[2];                // {scale, sum(p)}

  const int tid  = threadIdx.x;
  const int lane = tid & 31;
  const int wave = tid >> 5;
  const int b    = blockIdx.x;

  const float* __restrict__ sb = story + (size_t)b * (size_t)(S_ * D_);

  qs[tid] = query[(size_t)b * D_ + tid];
  ws[tid] = Wa_w[tid];
  const float bias = Wa_b[0];          // note: cancels in softmax, kept for fidelity

  // Stage one 16x512 f32 tile (32 KB): 4 b128 async issues per lane, coalesced.
  auto issue_tile = [&](int t, int which) {
    const float* g = sb + (size_t)t * (TS * D_);
    #pragma unroll
    for (int j = 0; j < 4; ++j) {
      int f   = tid + j * NTHREADS;    // float4 index in tile, 0..2047
      int row = f >> 7;                // 128 float4 per row
      int c4  = f & 127;
      uint32_t dst = (uint32_t)(uintptr_t)&buf[which][row * LSTRIDE + c4 * 4];
      async_copy_b128(dst, g + row * D_ + c4 * 4);
    }
  };

  float m    = -INFINITY;              // running max (maintained by wave 0 only)
  float lsum = 0.0f;                   // running normalizer (all threads, via LDS scalars)
#if USE_WMMA
  const int hi  = lane >> 4;           // lane group (K offset +2 in A/B frags)
  const int l16 = lane & 15;
  const int d0  = wave * 32;           // this wave's 32 output columns
  const int d1  = d0 + 16;
  v8f c0 = {0.f,0.f,0.f,0.f,0.f,0.f,0.f,0.f};
  v8f c1 = c0;
#else
  float accd = 0.0f;                   // fallback: thread owns output column d=tid
#endif

  issue_tile(0, 0);

  for (int t = 0; t < NT; ++t) {
    const int cur = t & 1;
    if (t + 1 < NT) { issue_tile(t + 1, cur ^ 1); wait_asynccnt<4>(); }
    else            { wait_asynccnt<0>(); }
    __syncthreads();                   // (1) tile t resident + all waves arrived

    const float* __restrict__ X = buf[cur];

    // ---- Phase A: scores. Wave w reduces row w: sum_d tanh(x+q[d])*Wa_w[d] ----
    {
      const float* xr = X + wave * LSTRIDE;
      float a = 0.0f;
      #pragma unroll
      for (int i = 0; i < 16; ++i) {
        int d = lane + i * 32;
        a += tanhf(xr[d] + qs[d]) * ws[d];
      }
      #pragma unroll
      for (int off = 16; off > 0; off >>= 1)
        a += __shfl_xor(a, off, 32);
      if (lane == 0) sc[wave] = a + bias;
    }
    __syncthreads();                   // (2) sc[] visible

    // ---- Phase B: wave 0 computes softmax weights once, publishes via LDS ----
    if (wave == 0) {
      const float s = sc[lane & 15];   // lanes 16-31 mirror lanes 0-15
      float tmax = s;
      #pragma unroll
      for (int off = 8; off > 0; off >>= 1)
        tmax = fmaxf(tmax, __shfl_xor(tmax, off, 32));
      const float mn    = fmaxf(m, tmax);
      const float scale = __expf(m - mn);    // exp(-inf)=0 handles first tile
      m = mn;
      const float pj = __expf(s - mn);
      float ps = pj;
      #pragma unroll
      for (int off = 8; off > 0; off >>= 1)
        ps += __shfl_xor(ps, off, 32);
      if (lane < 16) pb[lane] = pj;
      if (lane == 0) { scl[0] = scale; scl[1] = ps; }
    }
    __syncthreads();                   // (3) pb[], scl[] visible

    const float scale = scl[0];
    lsum = lsum * scale + scl[1];

#if USE_WMMA
    // ---- Phase C: acc[d] = acc[d]*scale + sum_s p[s]*X[s][d] via WMMA ----
    // A (16x4 f32, documented layout): A[m][k] = X[s=4kk+K][d0+m], K = vgpr + 2*hi.
    // B (4x16): p replicated across N -> every column of C equals acc chunk.
    // bf is a single aligned ds_load_b64 (r0 even) -> no register-array selects.
    #pragma unroll
    for (int i = 0; i < 8; ++i) { c0[i] *= scale; c1[i] *= scale; }
    #pragma unroll
    for (int kk = 0; kk < 4; ++kk) {
      const int r0 = 4 * kk + 2 * hi;
      const v2f bf = *(const v2f*)&pb[r0];
      v2f a0, a1;
      a0[0] = X[(r0    ) * LSTRIDE + d0 + l16];
      a0[1] = X[(r0 + 1) * LSTRIDE + d0 + l16];
      a1[0] = X[(r0    ) * LSTRIDE + d1 + l16];
      a1[1] = X[(r0 + 1) * LSTRIDE + d1 + l16];
      c0 = __builtin_amdgcn_wmma_f32_16x16x4_f32(false, a0, false, bf, (short)0,
                                                 c0, false, false);
      c1 = __builtin_amdgcn_wmma_f32_16x16x4_f32(false, a1, false, bf, (short)0,
                                                 c1, false, false);
    }
#else
    float s2 = 0.0f;
    #pragma unroll
    for (int j = 0; j < TS; ++j) s2 += pb[j] * X[j * LSTRIDE + tid];
    accd = accd * scale + s2;
#endif
    __syncthreads();                   // (4) protect buf[cur] before re-issue at t+2
  }

  // ---- Epilogue: out = acc / l ----
  const float inv = 1.0f / lsum;
  float* __restrict__ ob = out + (size_t)b * D_;
#if USE_WMMA
  // C layout: VGPR r, lanes 0-15 -> M=r (N=lane); lanes 16-31 -> M=8+r.
  // All N columns identical, so read column N=0: lanes 0 and 16 hold M=0..7 / 8..15.
  if (l16 == 0) {
    const int mo = hi ? 8 : 0;
    #pragma unroll
    for (int r = 0; r < 8; ++r) {
      ob[d0 + mo + r] = c0[r] * inv;
      ob[d1 + mo + r] = c1[r] * inv;
    }
  }
#else
  ob[tid] = accd * inv;
#endif
}

extern "C" void kernel_launch(void* const* d_in, const int* in_sizes, int n_in,
                              void* d_out, int out_size, void* d_ws, size_t ws_size,
                              hipStream_t stream) {
  (void)n_in; (void)out_size; (void)d_ws; (void)ws_size;
  const float* story = (const float*)d_in[0];
  const float* query = (const float*)d_in[1];
  const float* Wa_w  = (const float*)d_in[2];
  const float* Wa_b  = (const float*)d_in[3];
  float* out = (float*)d_out;
  const int B = in_sizes[1] / D_;      // query is [B, D]
  hipLaunchKernelGGL(memn2n_attn_kernel, dim3(B), dim3(NTHREADS), 0, stream,
                     story, query, Wa_w, Wa_b, out);
}